// Gaussian_neighbor_54391465836756
// MI455X (gfx1250) — compile-verified
//
#include <hip/hip_runtime.h>
#include <stdint.h>

// Problem constants (match reference setup_inputs)
constexpr int Bc    = 4;
constexpr int Lc    = 1024;
constexpr int Ac    = 5;
constexpr int KNNc  = 32;
constexpr int Kc    = 16;
constexpr int PAIRS = Ac * Ac;                 // 25
constexpr int EDGE_ITEMS = KNNc * PAIRS;       // 800  (k, i, j) tuples per (b,l)
constexpr int OUT_PER_BL = EDGE_ITEMS * Kc;    // 12800 outputs per (b,l)
constexpr int NBR_DW  = KNNc * 15;             // 480 neighbor-coord dwords
constexpr int MASK_DW = KNNc * Ac;             // 160 neighbor-mask dwords

__device__ __forceinline__ float fast_exp2(float x) {
#if __has_builtin(__builtin_amdgcn_exp2f)
    return __builtin_amdgcn_exp2f(x);   // raw v_exp_f32
#else
    return exp2f(x);
#endif
}

__global__ __launch_bounds__(256, 4)
void gauss_nbr_kernel(const float* __restrict__ X,
                      const int*   __restrict__ E_idx,
                      const float* __restrict__ mask_atoms,
                      const float* __restrict__ mask_attend,
                      const float* __restrict__ means,
                      const float* __restrict__ stds,
                      const float* __restrict__ mul,
                      const float* __restrict__ bias,
                      float* __restrict__ out)
{
    __shared__ float s_nbr[NBR_DW];     // gathered neighbor coords (32 rows x 15)
    __shared__ float s_mask[MASK_DW];   // gathered neighbor atom masks (32 x 5)
    __shared__ float s_ctr[16];         // center coords (15 used)
    __shared__ float s_mul[PAIRS];
    __shared__ float s_bias[PAIRS];
    __shared__ float s_a[Kc];           // (kz/std),      kz = sqrt(0.5*log2(e))
    __shared__ float s_b[Kc];           // (-kz*mean/std)
    __shared__ float s_lf[Kc];          // log2( 1/(sqrt(2*pi)*std) )
    __shared__ float s_att[KNNc];
    __shared__ float s_x[EDGE_ITEMS];   // masked affine distances

    const int tid = threadIdx.x;
    const int bl  = blockIdx.x;         // b*L + l
    const int b   = bl >> 10;           // L = 1024
    const long edgeBase = (long)bl * KNNc;

    // ---------------- phase 0: stage small params (regular loads) ------------
    if (tid < 15) s_ctr[tid] = X[(long)bl * 15 + tid];
    if (tid < PAIRS) { s_mul[tid] = mul[tid]; s_bias[tid] = bias[tid]; }
    if (tid < Kc) {
        // coef * exp(-0.5*z^2) == 2^( -(kz*z)^2 + log2(coef) )
        const float kz = sqrtf(0.5f * 1.44269504088896f);   // folds at compile time
        float sd  = fabsf(stds[tid]) + 0.01f;
        float inv = 1.0f / sd;
        s_a[tid]  = inv * kz;
        s_b[tid]  = -means[tid] * inv * kz;
        // log2(0.39894228/sd); sd >= 0.01 so argument is always positive/finite
        s_lf[tid] = log2f(0.39894228040143f * inv);   // 1/sqrt(2*3.1415926)
    }
    if (tid < KNNc) s_att[tid] = mask_attend[edgeBase + tid];

    // ---------------- phase 0b: async gather of neighbor rows into LDS -------
    // Generic LDS pointers have the LDS byte offset in their low 32 bits.
    const uint32_t nbr_lds0  = (uint32_t)(uintptr_t)(&s_nbr[0]);
    const uint32_t mask_lds0 = (uint32_t)(uintptr_t)(&s_mask[0]);

    for (int t = tid; t < NBR_DW + MASK_DW; t += 256) {
        const float* src;
        uint32_t ldsaddr;
        if (t < NBR_DW) {
            int k   = t / 15;
            int off = t - k * 15;
            int idx = E_idx[edgeBase + k];
            src     = X + ((long)(b * Lc + idx) * 15 + off);
            ldsaddr = nbr_lds0 + 4u * (uint32_t)t;
        } else {
            int t2  = t - NBR_DW;
            int k   = t2 / Ac;
            int off = t2 - k * Ac;
            int idx = E_idx[edgeBase + k];
            src     = mask_atoms + ((long)(b * Lc + idx) * Ac + off);
            ldsaddr = mask_lds0 + 4u * (uint32_t)t2;
        }
        uint64_t ga = (uint64_t)(uintptr_t)src;
        // CDNA5 async global->LDS copy (tracked by ASYNCcnt)
        asm volatile("global_load_async_to_lds_b32 %0, %1, off"
                     :: "v"(ldsaddr), "v"(ga) : "memory");
    }
    asm volatile("s_wait_asynccnt 0" ::: "memory");
    __syncthreads();

    // ---------------- phase 1: masked affine distances -> LDS ----------------
    for (int e = tid; e < EDGE_ITEMS; e += 256) {
        int k = e / PAIRS;
        int p = e - k * PAIRS;
        int i = p / Ac;
        int j = p - i * Ac;
        const float* nb = &s_nbr[k * 15 + i * 3];
        const float* cc = &s_ctr[j * 3];
        float dx = nb[0] - cc[0];
        float dy = nb[1] - cc[1];
        float dz = nb[2] - cc[2];
        float D  = sqrtf(fmaf(dx, dx, fmaf(dy, dy, dz * dz)));
        float m  = s_mask[k * Ac + i] * s_mask[k * Ac + j];
        s_x[e]   = fmaf(s_mul[p], D, s_bias[p]) * m;
    }
    __syncthreads();

    // ---------------- phase 2: 16-channel Gaussian RBF, b128 stores ----------
    float pa[Kc], pb[Kc], plf[Kc];
#pragma unroll
    for (int c = 0; c < Kc; ++c) { pa[c] = s_a[c]; pb[c] = s_b[c]; plf[c] = s_lf[c]; }

    float* outBL = out + (long)bl * OUT_PER_BL;
    for (int e = tid; e < EDGE_ITEMS; e += 256) {
        int   k   = e / PAIRS;
        float xv  = s_x[e];
        float att = s_att[k];
        float ov[Kc];
#pragma unroll
        for (int c = 0; c < Kc; ++c) {
            float z = fmaf(xv, pa[c], pb[c]);     // scaled by sqrt(0.5*log2e)
            float w = fmaf(-z, z, plf[c]);        // -z^2 + log2(coef); neg is free
            ov[c]   = fast_exp2(w) * att;         // v_exp_f32 + one mul
        }
        float4* dst = (float4*)(outBL + (long)e * Kc);  // 64B-aligned
        dst[0] = make_float4(ov[0],  ov[1],  ov[2],  ov[3]);
        dst[1] = make_float4(ov[4],  ov[5],  ov[6],  ov[7]);
        dst[2] = make_float4(ov[8],  ov[9],  ov[10], ov[11]);
        dst[3] = make_float4(ov[12], ov[13], ov[14], ov[15]);
    }
}

extern "C" void kernel_launch(void* const* d_in, const int* in_sizes, int n_in,
                              void* d_out, int out_size, void* d_ws, size_t ws_size,
                              hipStream_t stream) {
    const float* X     = (const float*)d_in[0];
    const int*   E     = (const int*)  d_in[1];
    const float* ma    = (const float*)d_in[2];
    const float* matt  = (const float*)d_in[3];
    const float* means = (const float*)d_in[4];
    const float* stds  = (const float*)d_in[5];
    const float* mulp  = (const float*)d_in[6];
    const float* biasp = (const float*)d_in[7];
    float* out = (float*)d_out;

    dim3 grid(Bc * Lc);   // 4096 blocks, one per (b,l)
    dim3 block(256);      // 8 wave32 per block
    gauss_nbr_kernel<<<grid, block, 0, stream>>>(X, E, ma, matt, means, stds,
                                                 mulp, biasp, out);
}